// VNFrameEstimator_1640677507534
// MI455X (gfx1250) — compile-verified
//
#include <hip/hip_runtime.h>
#include <math.h>

typedef __attribute__((ext_vector_type(2))) float v2f;
typedef __attribute__((ext_vector_type(8))) float v8f;

#define VN_SLOPE 0.2f
#define VN_EPS   1e-6f

// ---------------------------------------------------------------------------
// Farthest point sampling: one block per batch, dist array kept in LDS.
// Matches jax scan: fidx[i] = far_before_step_i, argmax keeps lowest index.
// ---------------------------------------------------------------------------
__global__ void fps_kernel(const float* __restrict__ xyz, int* __restrict__ fidx,
                           int N, int npoint) {
  const int T = 512;
  int b = blockIdx.x, t = threadIdx.x;
  __shared__ float dist[8192];
  __shared__ float rv[512];
  __shared__ int   ri[512];
  const float* xb = xyz + (size_t)b * 3 * N;
  for (int n = t; n < N; n += T) dist[n] = 1e10f;
  __syncthreads();
  int far = 0;
  for (int i = 0; i < npoint; ++i) {
    if (t == 0) fidx[b * npoint + i] = far;
    float cx = xb[far], cy = xb[N + far], cz = xb[2 * N + far];
    float best = -1.0f; int bi = 0x7fffffff;
    for (int n = t; n < N; n += T) {
      float dx = xb[n] - cx, dy = xb[N + n] - cy, dz = xb[2 * N + n] - cz;
      float d = dx * dx + dy * dy + dz * dz;
      float dn = fminf(dist[n], d);
      dist[n] = dn;
      if (dn > best) { best = dn; bi = n; }   // strict > keeps lowest index
    }
    rv[t] = best; ri[t] = bi;
    __syncthreads();
    for (int s = T / 2; s > 0; s >>= 1) {
      if (t < s) {
        if (rv[t + s] > rv[t] || (rv[t + s] == rv[t] && ri[t + s] < ri[t])) {
          rv[t] = rv[t + s]; ri[t] = ri[t + s];
        }
      }
      __syncthreads();
    }
    far = ri[0];
    __syncthreads();
  }
}

// Gather sampled centers into (B,3,S) layout.
__global__ void gather_xyz_kernel(const float* __restrict__ src,
                                  const int* __restrict__ fidx,
                                  float* __restrict__ dst,
                                  int N, int S, int B) {
  int i = blockIdx.x * blockDim.x + threadIdx.x;
  if (i >= B * 3 * S) return;
  int s = i % S;
  int d = (i / S) % 3;
  int b = i / (3 * S);
  dst[i] = src[(size_t)b * 3 * N + (size_t)d * N + fidx[b * S + s]];
}

// ---------------------------------------------------------------------------
// KNN: one block per (b, center). Distances in LDS, k iterative argmin picks.
// ---------------------------------------------------------------------------
__global__ void knn_kernel(const float* __restrict__ xyz, const int* __restrict__ fidx,
                           int* __restrict__ nidx, int N, int S, int K) {
  const int T = 256;
  int bs = blockIdx.x;
  int b = bs / S;
  int t = threadIdx.x;
  __shared__ float dist[8192];
  __shared__ float rv[256];
  __shared__ int   ri[256];
  const float* xb = xyz + (size_t)b * 3 * N;
  int ci = fidx[bs];
  float cx = xb[ci], cy = xb[N + ci], cz = xb[2 * N + ci];
  for (int n = t; n < N; n += T) {
    float dx = xb[n] - cx, dy = xb[N + n] - cy, dz = xb[2 * N + n] - cz;
    dist[n] = dx * dx + dy * dy + dz * dz;
  }
  __syncthreads();
  for (int j = 0; j < K; ++j) {
    float best = 3e38f; int bi = 0x7fffffff;
    for (int n = t; n < N; n += T) {
      if (dist[n] < best) { best = dist[n]; bi = n; }
    }
    rv[t] = best; ri[t] = bi;
    __syncthreads();
    for (int s = T / 2; s > 0; s >>= 1) {
      if (t < s) {
        if (rv[t + s] < rv[t] || (rv[t + s] == rv[t] && ri[t + s] < ri[t])) {
          rv[t] = rv[t + s]; ri[t] = ri[t + s];
        }
      }
      __syncthreads();
    }
    if (t == 0) {
      nidx[bs * K + j] = ri[0];
      dist[ri[0]] = 3e38f;
    }
    __syncthreads();
  }
}

// ---------------------------------------------------------------------------
// WMMA fp32 GEMM:  Y[m][n] = sum_k X[m][k] * W[n][k]
// Each wave owns a 64x16 output region (4 stacked 16x16 tiles). K is consumed
// in blocks of 16 (4 WMMA-steps): all 20 fragment loads of a block are issued
// as one batch before its 16 WMMAs, ordered to match consumption, so the
// compiler can use one load clause + progressively decreasing partial
// s_wait_loadcnt instead of a full drain per matrix op. All hot-path K values
// (64/128/256, and 128 for K=129) are multiples of 16; a generic 4-step loop
// plus a guarded tail covers the rest.
//
// A layout (32-bit 16x4): lanes 0-15 rows M=0..15 with K=0,1; lanes 16-31 K=2,3.
// B layout mirrors K split: V0 = {K0 | K2}, V1 = {K1 | K3}, N striped on lanes.
// C/D: VGPR j -> rows j (lanes 0-15) and j+8 (lanes 16-31), N = lane%16.
// ---------------------------------------------------------------------------
__global__ void gemm_vn_kernel(const float* __restrict__ X, const float* __restrict__ W,
                               float* __restrict__ Y, int Mrows, int Kpad, int Kreal,
                               int Cout) {
  int wave = (blockIdx.x * blockDim.x + threadIdx.x) >> 5;
  int Mt4 = Mrows >> 6;                 // 64-row macro-tiles
  int Nt  = Cout >> 4;
  if (wave >= Mt4 * Nt) return;         // wave-uniform: WMMA only with EXEC all-1
  int mt = wave % Mt4, nt = wave / Mt4;
  int lane = threadIdx.x & 31;
  int half = lane >> 4;                 // 0: K={c0,c0+1}, 1: K={c0+2,c0+3}
  int l16  = lane & 15;

  const float* A0 = X + (size_t)(mt * 64 + l16) * Kpad + half * 2;
  const size_t rstep = (size_t)16 * Kpad;          // 16 rows
  int n = nt * 16 + l16;
  const float* Wn = W + (size_t)n * Kreal + half * 2;

  v8f acc0 = {}, acc1 = {}, acc2 = {}, acc3 = {};
  int Kmain = Kreal & ~3;
  int Kblk  = Kmain & ~15;              // multiple-of-16 part (all hot paths)

  int kk = 0;
  for (; kk < Kblk; kk += 16) {
    // ---- batched loads: 4 B-fragments + 16 A-fragments ----------------
    v2f bf[4], af[4][4];                // af[step][subtile]
#pragma unroll
    for (int s = 0; s < 4; ++s) {
      int kc = kk + 4 * s;
      bf[s][0]    = Wn[kc];               bf[s][1]    = Wn[kc + 1];
      af[s][0][0] = A0[kc];               af[s][0][1] = A0[kc + 1];
      af[s][1][0] = A0[rstep + kc];       af[s][1][1] = A0[rstep + kc + 1];
      af[s][2][0] = A0[2 * rstep + kc];   af[s][2][1] = A0[2 * rstep + kc + 1];
      af[s][3][0] = A0[3 * rstep + kc];   af[s][3][1] = A0[3 * rstep + kc + 1];
    }
    // ---- 16 WMMAs consuming the batch in load order -------------------
#pragma unroll
    for (int s = 0; s < 4; ++s) {
      acc0 = __builtin_amdgcn_wmma_f32_16x16x4_f32(false, af[s][0], false, bf[s], (short)0, acc0, false, false);
      acc1 = __builtin_amdgcn_wmma_f32_16x16x4_f32(false, af[s][1], false, bf[s], (short)0, acc1, false, false);
      acc2 = __builtin_amdgcn_wmma_f32_16x16x4_f32(false, af[s][2], false, bf[s], (short)0, acc2, false, false);
      acc3 = __builtin_amdgcn_wmma_f32_16x16x4_f32(false, af[s][3], false, bf[s], (short)0, acc3, false, false);
    }
  }

  for (; kk < Kmain; kk += 4) {         // generic 4-step remainder (cold path)
    v2f bv, a0, a1, a2, a3;
    bv[0] = Wn[kk];             bv[1] = Wn[kk + 1];
    a0[0] = A0[kk];             a0[1] = A0[kk + 1];
    a1[0] = A0[rstep + kk];     a1[1] = A0[rstep + kk + 1];
    a2[0] = A0[2 * rstep + kk]; a2[1] = A0[2 * rstep + kk + 1];
    a3[0] = A0[3 * rstep + kk]; a3[1] = A0[3 * rstep + kk + 1];
    acc0 = __builtin_amdgcn_wmma_f32_16x16x4_f32(false, a0, false, bv, (short)0, acc0, false, false);
    acc1 = __builtin_amdgcn_wmma_f32_16x16x4_f32(false, a1, false, bv, (short)0, acc1, false, false);
    acc2 = __builtin_amdgcn_wmma_f32_16x16x4_f32(false, a2, false, bv, (short)0, acc2, false, false);
    acc3 = __builtin_amdgcn_wmma_f32_16x16x4_f32(false, a3, false, bv, (short)0, acc3, false, false);
  }

  if (Kmain < Kreal) {                  // single guarded tail step (Kreal % 4 != 0)
    int c0 = Kmain + half * 2;
    v2f bv, a0, a1, a2, a3;
    bv[0] = (c0     < Kreal) ? Wn[Kmain]     : 0.0f;
    bv[1] = (c0 + 1 < Kreal) ? Wn[Kmain + 1] : 0.0f;
    a0[0] = A0[Kmain];             a0[1] = A0[Kmain + 1];   // X zero-padded to Kpad
    a1[0] = A0[rstep + Kmain];     a1[1] = A0[rstep + Kmain + 1];
    a2[0] = A0[2 * rstep + Kmain]; a2[1] = A0[2 * rstep + Kmain + 1];
    a3[0] = A0[3 * rstep + Kmain]; a3[1] = A0[3 * rstep + Kmain + 1];
    acc0 = __builtin_amdgcn_wmma_f32_16x16x4_f32(false, a0, false, bv, (short)0, acc0, false, false);
    acc1 = __builtin_amdgcn_wmma_f32_16x16x4_f32(false, a1, false, bv, (short)0, acc1, false, false);
    acc2 = __builtin_amdgcn_wmma_f32_16x16x4_f32(false, a2, false, bv, (short)0, acc2, false, false);
    acc3 = __builtin_amdgcn_wmma_f32_16x16x4_f32(false, a3, false, bv, (short)0, acc3, false, false);
  }

  float* Yb = Y + (size_t)(mt * 64 + half * 8) * Cout + nt * 16 + l16;
#pragma unroll
  for (int j = 0; j < 8; ++j) Yb[(size_t)j * Cout] = acc0[j];
  Yb += (size_t)16 * Cout;
#pragma unroll
  for (int j = 0; j < 8; ++j) Yb[(size_t)j * Cout] = acc1[j];
  Yb += (size_t)16 * Cout;
#pragma unroll
  for (int j = 0; j < 8; ++j) Yb[(size_t)j * Cout] = acc2[j];
  Yb += (size_t)16 * Cout;
#pragma unroll
  for (int j = 0; j < 8; ++j) Yb[(size_t)j * Cout] = acc3[j];
}

// ---------------------------------------------------------------------------
// VN batch-norm on vector norms (per point p, channel o; rows 3p..3p+2).
// bn params stacked rows: [gamma, beta, running_mean, running_var].
// ---------------------------------------------------------------------------
__global__ void vn_bn_kernel(float* __restrict__ Y, const float* __restrict__ bn,
                             int npts, int C) {
  int i = blockIdx.x * blockDim.x + threadIdx.x;
  if (i >= npts * C) return;
  int p = i / C, o = i % C;
  size_t base = (size_t)p * 3 * C + o;
  float x0 = Y[base], x1 = Y[base + C], x2 = Y[base + 2 * (size_t)C];
  float norm = sqrtf(x0 * x0 + x1 * x1 + x2 * x2) + VN_EPS;
  float gamma = bn[o], beta = bn[C + o], rm = bn[2 * C + o], rvv = bn[3 * C + o];
  float nbn = gamma * (norm - rm) * rsqrtf(rvv + 1e-5f) + beta;
  float sc = nbn / norm;
  Y[base] = x0 * sc;
  Y[base + C] = x1 * sc;
  Y[base + 2 * (size_t)C] = x2 * sc;
}

// VN leaky relu: reads Y (x) and D (direction d), writes result over Y in place.
__global__ void vn_leaky_kernel(float* __restrict__ Y, const float* __restrict__ D,
                                int npts, int C) {
  int i = blockIdx.x * blockDim.x + threadIdx.x;
  if (i >= npts * C) return;
  int p = i / C, o = i % C;
  size_t base = (size_t)p * 3 * C + o;
  float x0 = Y[base], x1 = Y[base + C], x2 = Y[base + 2 * (size_t)C];
  float d0 = D[base], d1 = D[base + C], d2 = D[base + 2 * (size_t)C];
  float dot = x0 * d0 + x1 * d1 + x2 * d2;
  float dsq = d0 * d0 + d1 * d1 + d2 * d2;
  float o0, o1, o2;
  if (dot >= 0.0f) {
    o0 = x0; o1 = x1; o2 = x2;           // slope*x + (1-slope)*x
  } else {
    float c = dot / (dsq + VN_EPS);
    o0 = VN_SLOPE * x0 + (1.0f - VN_SLOPE) * (x0 - c * d0);
    o1 = VN_SLOPE * x1 + (1.0f - VN_SLOPE) * (x1 - c * d1);
    o2 = VN_SLOPE * x2 + (1.0f - VN_SLOPE) * (x2 - c * d2);
  }
  Y[base] = o0;
  Y[base + C] = o1;
  Y[base + 2 * (size_t)C] = o2;
}

// ---------------------------------------------------------------------------
// Grouping for SA1: X0 rows = (point*3+comp), Kpad=4 cols (col0 = rel coord).
// ---------------------------------------------------------------------------
__global__ void group1_kernel(const float* __restrict__ x, const int* __restrict__ fidx,
                              const int* __restrict__ nidx, float* __restrict__ X0,
                              int g0, int Gc, int N, int S, int Ksz) {
  int p = blockIdx.x * blockDim.x + threadIdx.x;
  if (p >= Gc * Ksz) return;
  int gl = p / Ksz, k = p % Ksz;
  int gg = g0 + gl;
  int b = gg / S;
  const float* xb = x + (size_t)b * 3 * N;
  int ni = nidx[gg * Ksz + k];
  int ci = fidx[gg];
#pragma unroll
  for (int d = 0; d < 3; ++d) {
    float rel = xb[(size_t)d * N + ni] - xb[(size_t)d * N + ci];
    size_t row = (size_t)(p * 3 + d) * 4;
    X0[row + 0] = rel;
    X0[row + 1] = 0.0f;
    X0[row + 2] = 0.0f;
    X0[row + 3] = 0.0f;
  }
}

// Grouping for SA2: channel 0 = rel xyz, 1..128 = gathered feat1, 129..131 = pad.
__global__ void group2_kernel(const float* __restrict__ xyz1, const float* __restrict__ feat1,
                              const int* __restrict__ fidx, const int* __restrict__ nidx,
                              float* __restrict__ X, int g0, int Gc, int N, int S,
                              int Ksz, int Kpad) {
  int i = blockIdx.x * blockDim.x + threadIdx.x;
  if (i >= Gc * Ksz * Kpad) return;
  int c = i % Kpad;
  int p = i / Kpad;
  int gl = p / Ksz, k = p % Ksz;
  int gg = g0 + gl;
  int b = gg / S;
  int ni = nidx[gg * Ksz + k];
  int ci = fidx[gg];
#pragma unroll
  for (int comp = 0; comp < 3; ++comp) {
    float v;
    if (c == 0) {
      v = xyz1[(size_t)b * 3 * N + (size_t)comp * N + ni] -
          xyz1[(size_t)b * 3 * N + (size_t)comp * N + ci];
    } else if (c <= 128) {
      v = feat1[(((size_t)b * 128 + (c - 1)) * 256 + ni) * 3 + comp];
    } else {
      v = 0.0f;
    }
    X[((size_t)p * 3 + comp) * Kpad + c] = v;
  }
}

// vn_max_pool over the neighbor dim of one chunk; writes (B,C,S,3) layout.
__global__ void pool_kernel(const float* __restrict__ Xc, float* __restrict__ out,
                            int g0, int Gc, int Ksz, int C, int S) {
  int i = blockIdx.x * blockDim.x + threadIdx.x;
  if (i >= Gc * C) return;
  int gl = i / C, o = i % C;
  float best = -1.0f; int bk = 0;
  for (int k = 0; k < Ksz; ++k) {
    size_t base = (size_t)((gl * Ksz + k) * 3) * C + o;
    float a0 = Xc[base], a1 = Xc[base + C], a2 = Xc[base + 2 * (size_t)C];
    float n2 = a0 * a0 + a1 * a1 + a2 * a2;
    if (n2 > best) { best = n2; bk = k; }
  }
  int gg = g0 + gl;
  int b = gg / S, s = gg % S;
  size_t base = (size_t)((gl * Ksz + bk) * 3) * C + o;
  size_t ob = (((size_t)b * C + o) * S + s) * 3;
  out[ob + 0] = Xc[base];
  out[ob + 1] = Xc[base + C];
  out[ob + 2] = Xc[base + 2 * (size_t)C];
}

// ---------------------------------------------------------------------------
// Final: pool v_feat2 over points, project with W_pred, Gram-Schmidt frame.
// One block per batch, 256 threads (one per channel).
// ---------------------------------------------------------------------------
__global__ void final_kernel(const float* __restrict__ vfeat2, const float* __restrict__ Wp,
                             float* __restrict__ R) {
  int b = blockIdx.x, t = threadIdx.x;
  __shared__ float pooled[256][3];
  __shared__ float Msh[2][3];
  const float* vb = vfeat2 + ((size_t)b * 256 + t) * 128 * 3;
  float best = -1.0f; int bs_ = 0;
  for (int s = 0; s < 128; ++s) {
    float a0 = vb[s * 3], a1 = vb[s * 3 + 1], a2 = vb[s * 3 + 2];
    float n2 = a0 * a0 + a1 * a1 + a2 * a2;
    if (n2 > best) { best = n2; bs_ = s; }
  }
  pooled[t][0] = vb[bs_ * 3];
  pooled[t][1] = vb[bs_ * 3 + 1];
  pooled[t][2] = vb[bs_ * 3 + 2];
  __syncthreads();
  if (t < 6) {
    int r = t / 3, comp = t % 3;
    float acc = 0.0f;
    for (int o = 0; o < 256; ++o) acc += Wp[r * 256 + o] * pooled[o][comp];
    Msh[r][comp] = acc;
  }
  __syncthreads();
  if (t == 0) {
    float v1[3] = {Msh[0][0], Msh[0][1], Msh[0][2]};
    float v2[3] = {Msh[1][0], Msh[1][1], Msh[1][2]};
    float n1 = sqrtf(v1[0] * v1[0] + v1[1] * v1[1] + v1[2] * v1[2]);
    float inv1 = 1.0f / fmaxf(n1, 1e-6f);
    float u1[3] = {v1[0] * inv1, v1[1] * inv1, v1[2] * inv1};
    float dp = u1[0] * v2[0] + u1[1] * v2[1] + u1[2] * v2[2];
    float u2[3] = {v2[0] - dp * u1[0], v2[1] - dp * u1[1], v2[2] - dp * u1[2]};
    float n2 = sqrtf(u2[0] * u2[0] + u2[1] * u2[1] + u2[2] * u2[2]);
    float inv2 = 1.0f / fmaxf(n2, 1e-6f);
    u2[0] *= inv2; u2[1] *= inv2; u2[2] *= inv2;
    float u3[3] = {u1[1] * u2[2] - u1[2] * u2[1],
                   u1[2] * u2[0] - u1[0] * u2[2],
                   u1[0] * u2[1] - u1[1] * u2[0]};
    float* Rb = R + b * 9;
    for (int i = 0; i < 3; ++i) {
      Rb[i * 3 + 0] = u1[i];
      Rb[i * 3 + 1] = u2[i];
      Rb[i * 3 + 2] = u3[i];
    }
  }
}

// ---------------------------------------------------------------------------
static inline int ceil_div(int a, int b) { return (a + b - 1) / b; }

static void run_gemm(const float* X, const float* W, float* Y, int M, int Kpad,
                     int Kreal, int Cout, hipStream_t st) {
  int tiles = (M / 64) * (Cout / 16);   // one 64x16 region per wave
  int blocks = ceil_div(tiles, 8);      // 8 waves per 256-thread block
  gemm_vn_kernel<<<blocks, 256, 0, st>>>(X, W, Y, M, Kpad, Kreal, Cout);
}

extern "C" void kernel_launch(void* const* d_in, const int* in_sizes, int n_in,
                              void* d_out, int out_size, void* d_ws, size_t ws_size,
                              hipStream_t stream) {
  (void)in_sizes; (void)n_in; (void)out_size; (void)ws_size;
  const float* x       = (const float*)d_in[0];
  const float* sa1_W1  = (const float*)d_in[1];
  const float* sa1_D1  = (const float*)d_in[2];
  const float* sa1_bn1 = (const float*)d_in[3];
  const float* sa1_W2  = (const float*)d_in[4];
  const float* sa1_D2  = (const float*)d_in[5];
  const float* sa1_bn2 = (const float*)d_in[6];
  const float* sa2_W1  = (const float*)d_in[7];
  const float* sa2_D1  = (const float*)d_in[8];
  const float* sa2_bn1 = (const float*)d_in[9];
  const float* sa2_W2  = (const float*)d_in[10];
  const float* sa2_D2  = (const float*)d_in[11];
  const float* sa2_bn2 = (const float*)d_in[12];
  const float* W_pred  = (const float*)d_in[13];

  const int B = 16, N = 8192;
  const int S1 = 256, K1 = 32;   // SA1 npoint/nsample
  const int S2 = 128, K2 = 16;   // SA2 npoint/nsample

  // Output regions (reference return order, all float32)
  float* out = (float*)d_out;
  float* R      = out;                                  // 16*3*3        = 144
  float* vfeat2 = R + 144;                              // 16*256*128*3  = 1572864
  float* xyz2   = vfeat2 + (size_t)16 * 256 * 128 * 3;  // 16*3*128      = 6144
  float* feat1  = xyz2 + (size_t)16 * 3 * 128;          // 16*128*256*3  = 1572864
  float* xyz1   = feat1 + (size_t)16 * 128 * 256 * 3;   // 16*3*256      = 12288

  // Workspace layout (256B-aligned regions)
  char* ws = (char*)d_ws;
  size_t off = 0;
  auto take = [&](size_t bytes) {
    void* p = ws + off;
    off += (bytes + 255) & ~(size_t)255;
    return p;
  };
  int*   fidx1 = (int*)  take((size_t)B * S1 * sizeof(int));
  int*   nidx1 = (int*)  take((size_t)B * S1 * K1 * sizeof(int));
  int*   fidx2 = (int*)  take((size_t)B * S2 * sizeof(int));
  int*   nidx2 = (int*)  take((size_t)B * S2 * K2 * sizeof(int));
  const size_t BUF_ELEMS = (size_t)49152 * 128;   // = 24576*256 too
  float* bufA  = (float*)take(BUF_ELEMS * sizeof(float));
  float* bufB  = (float*)take(BUF_ELEMS * sizeof(float));

  // ---- SA1 sampling / grouping indices --------------------------------
  fps_kernel<<<B, 512, 0, stream>>>(x, fidx1, N, S1);
  gather_xyz_kernel<<<ceil_div(B * 3 * S1, 256), 256, 0, stream>>>(x, fidx1, xyz1, N, S1, B);
  knn_kernel<<<B * S1, 256, 0, stream>>>(x, fidx1, nidx1, N, S1, K1);

  // ---- SA1 VN layers, chunked over groups -----------------------------
  {
    const int G = B * S1;        // 4096 groups
    const int GC = 512;          // groups per chunk
    const int PTS = GC * K1;     // 16384 points per chunk
    const int M = PTS * 3;       // 49152 rows
    for (int g0 = 0; g0 < G; g0 += GC) {
      group1_kernel<<<ceil_div(GC * K1, 256), 256, 0, stream>>>(
          x, fidx1, nidx1, bufA, g0, GC, N, S1, K1);
      run_gemm(bufA, sa1_W1, bufB, M, 4, 1, 64, stream);              // lin1
      vn_bn_kernel<<<ceil_div(PTS * 64, 256), 256, 0, stream>>>(bufB, sa1_bn1, PTS, 64);
      run_gemm(bufB, sa1_D1, bufA, M, 64, 64, 64, stream);            // d = D1*y
      vn_leaky_kernel<<<ceil_div(PTS * 64, 256), 256, 0, stream>>>(bufB, bufA, PTS, 64);
      run_gemm(bufB, sa1_W2, bufA, M, 64, 64, 128, stream);           // lin2
      vn_bn_kernel<<<ceil_div(PTS * 128, 256), 256, 0, stream>>>(bufA, sa1_bn2, PTS, 128);
      run_gemm(bufA, sa1_D2, bufB, M, 128, 128, 128, stream);         // d = D2*y
      vn_leaky_kernel<<<ceil_div(PTS * 128, 256), 256, 0, stream>>>(bufA, bufB, PTS, 128);
      pool_kernel<<<ceil_div(GC * 128, 256), 256, 0, stream>>>(
          bufA, feat1, g0, GC, K1, 128, S1);
    }
  }

  // ---- SA2 sampling / grouping indices (on xyz1) ----------------------
  fps_kernel<<<B, 512, 0, stream>>>(xyz1, fidx2, S1, S2);
  gather_xyz_kernel<<<ceil_div(B * 3 * S2, 256), 256, 0, stream>>>(xyz1, fidx2, xyz2, S1, S2, B);
  knn_kernel<<<B * S2, 256, 0, stream>>>(xyz1, fidx2, nidx2, S1, S2, K2);

  // ---- SA2 VN layers, chunked over groups -----------------------------
  {
    const int G = B * S2;        // 2048 groups
    const int GC = 512;
    const int PTS = GC * K2;     // 8192 points per chunk
    const int M = PTS * 3;       // 24576 rows
    const int KP = 132;          // 129 channels padded to multiple of 4
    for (int g0 = 0; g0 < G; g0 += GC) {
      group2_kernel<<<ceil_div(GC * K2 * KP, 256), 256, 0, stream>>>(
          xyz1, feat1, fidx2, nidx2, bufA, g0, GC, S1, S2, K2, KP);
      run_gemm(bufA, sa2_W1, bufB, M, KP, 129, 128, stream);          // lin1
      vn_bn_kernel<<<ceil_div(PTS * 128, 256), 256, 0, stream>>>(bufB, sa2_bn1, PTS, 128);
      run_gemm(bufB, sa2_D1, bufA, M, 128, 128, 128, stream);
      vn_leaky_kernel<<<ceil_div(PTS * 128, 256), 256, 0, stream>>>(bufB, bufA, PTS, 128);
      run_gemm(bufB, sa2_W2, bufA, M, 128, 128, 256, stream);         // lin2
      vn_bn_kernel<<<ceil_div(PTS * 256, 256), 256, 0, stream>>>(bufA, sa2_bn2, PTS, 256);
      run_gemm(bufA, sa2_D2, bufB, M, 256, 256, 256, stream);
      vn_leaky_kernel<<<ceil_div(PTS * 256, 256), 256, 0, stream>>>(bufA, bufB, PTS, 256);
      pool_kernel<<<ceil_div(GC * 256, 256), 256, 0, stream>>>(
          bufA, vfeat2, g0, GC, K2, 256, S2);
    }
  }

  // ---- Final pooled projection + Gram-Schmidt frame -------------------
  final_kernel<<<B, 256, 0, stream>>>(vfeat2, W_pred, R);
}